// CostVolumeConcat_19353122635920
// MI455X (gfx1250) — compile-verified
//
#include <hip/hip_runtime.h>

// Problem constants (from the reference):
//   B=1, C=16, H=128, W=512, D = 192/2 = 96
//   out shape (1, 2C=32, D=96, H=128, W=512), fp32
#define CV_W 512
#define CV_H 128
#define CV_C 16
#define CV_D 96

typedef __attribute__((ext_vector_type(4))) float v4f;

// Types matching the async-to-LDS builtin's expected parameters:
// arg0: __attribute__((vector_size(16))) int __device__ *   (AS1, global)
// arg1: presumably same pointee in AS3 (LDS)
typedef int b128_t __attribute__((vector_size(16)));
typedef __attribute__((address_space(1))) b128_t* gptr_b128;
typedef __attribute__((address_space(3))) b128_t* lptr_b128;

__global__ __launch_bounds__(128) void cost_volume_concat_kernel(
    const float* __restrict__ un_l,
    const float* __restrict__ un_r,
    float* __restrict__ out)
{
    __shared__ float row[CV_W];   // one input row, 2 KB

    const int t  = threadIdx.x;            // 0..127
    const int h  = blockIdx.x & (CV_H - 1);
    const int ch = blockIdx.x >> 7;        // 0..31 (output channel)
    const bool is_left = (ch < CV_C);

    const float* src = is_left
        ? (un_l + ((size_t)ch * CV_H + h) * CV_W)
        : (un_r + ((size_t)(ch - CV_C) * CV_H + h) * CV_W);

    // ---- Stage the 2 KB row into LDS (async path on CDNA5) ----
#if __has_builtin(__builtin_amdgcn_global_load_async_to_lds_b128)
    __builtin_amdgcn_global_load_async_to_lds_b128(
        (gptr_b128)(src + 4 * t),
        (lptr_b128)(&row[4 * t]),
        /*offset=*/0, /*cpol=*/0);
  #if __has_builtin(__builtin_amdgcn_s_wait_asynccnt)
    __builtin_amdgcn_s_wait_asynccnt(0);
  #else
    asm volatile("s_wait_asynccnt 0" ::: "memory");
  #endif
#else
    // Fallback: plain b128 load + LDS store
    *(v4f*)(&row[4 * t]) = *(const v4f*)(src + 4 * t);
#endif
    __syncthreads();

    // Output base for (ch, d=0, h, w=4t); stride between d-planes = H*W.
    float* outp = out + (size_t)ch * (CV_D * CV_H * CV_W)
                      + (size_t)h * CV_W + 4 * t;

    if (is_left) {
        // Same row for all 96 disparities: read LDS once, stream 96 NT stores.
        v4f v = *(const v4f*)(&row[4 * t]);
        #pragma unroll 8
        for (int d = 0; d < CV_D; ++d) {
            __builtin_nontemporal_store(v, (v4f*)outp);
            outp += CV_H * CV_W;
        }
    } else {
        // Rotated row: out[w] = row[(w - d) mod 512]. Store addresses stay
        // b128-aligned; the rotation goes into the LDS read index
        // (consecutive lanes read LDS words 4 apart).
        const int w0 = 4 * t;
        #pragma unroll 4
        for (int d = 0; d < CV_D; ++d) {
            v4f v;
            v.x = row[(w0 + 0 - d) & (CV_W - 1)];
            v.y = row[(w0 + 1 - d) & (CV_W - 1)];
            v.z = row[(w0 + 2 - d) & (CV_W - 1)];
            v.w = row[(w0 + 3 - d) & (CV_W - 1)];
            __builtin_nontemporal_store(v, (v4f*)outp);
            outp += CV_H * CV_W;
        }
    }
}

extern "C" void kernel_launch(void* const* d_in, const int* in_sizes, int n_in,
                              void* d_out, int out_size, void* d_ws, size_t ws_size,
                              hipStream_t stream) {
    const float* un_l = (const float*)d_in[0];  // (1,16,128,512) fp32
    const float* un_r = (const float*)d_in[1];  // (1,16,128,512) fp32
    float* out = (float*)d_out;                 // (1,32,96,128,512) fp32

    // One block per (output channel, h) pair: 32 * 128 = 4096 blocks,
    // 128 threads (4 wave32) each; each block writes 96 rows of 512 floats.
    dim3 grid(32 * CV_H);
    dim3 block(128);
    cost_volume_concat_kernel<<<grid, block, 0, stream>>>(un_l, un_r, out);
}